// LocationSensitiveAttention_38680475468125
// MI455X (gfx1250) — compile-verified
//
#include <hip/hip_runtime.h>
#include <math.h>

#define B_    64
#define V_    2048
#define HID_  1024
#define EMB_  512
#define ATTN_ 128
#define FILT_ 32
#define KERN_ 31
#define PAD_  15
#define KTOT  (EMB_ + FILT_)   // 544
#define KSTEPS (KTOT / 4)      // 136
#define MTILE 64
#define LDSS  548              // row stride (floats): mult of 4 -> 16B aligned rows;
                               // 548 % 64 = 36 -> no bank conflicts among 16-lane A reads

typedef __attribute__((ext_vector_type(2))) float v2f;
typedef __attribute__((ext_vector_type(8))) float v8f;
typedef __attribute__((ext_vector_type(4))) int   v4i;

#if defined(__AMDGCN__) && __has_builtin(__builtin_amdgcn_global_load_async_to_lds_b128) && \
    __has_builtin(__builtin_amdgcn_s_wait_asynccnt)
#define USE_ASYNC_LDS 1
typedef __attribute__((address_space(1))) v4i* gv4i_ptr;   // global
typedef __attribute__((address_space(3))) v4i* lv4i_ptr;   // LDS
#endif

// ---------------------------------------------------------------------------
// 1) Location conv: (B,2,V) * (FILT,2,KERN) -> convT[b][v][f]
// ---------------------------------------------------------------------------
__global__ void conv_loc_kernel(const float* __restrict__ la,
                                const float* __restrict__ Wconv,
                                float* __restrict__ convT) {
    __shared__ float wc[FILT_ * 2 * KERN_];          // 1984 floats
    __shared__ float slab[2][256 + KERN_ - 1];       // 2 x 286 halo
    int b = blockIdx.x >> 3;
    int v0 = (blockIdx.x & 7) * 256;
    int tid = threadIdx.x;

    for (int i = tid; i < FILT_ * 2 * KERN_; i += 256) wc[i] = Wconv[i];
    for (int c = 0; c < 2; ++c)
        for (int i = tid; i < 256 + KERN_ - 1; i += 256) {
            int idx = v0 + i - PAD_;
            slab[c][i] = (idx >= 0 && idx < V_) ? la[(b * 2 + c) * V_ + idx] : 0.f;
        }
    __syncthreads();

    int v = v0 + tid;
    float* dst = convT + ((size_t)b * V_ + v) * FILT_;
    for (int f = 0; f < FILT_; ++f) {
        const float* w0 = &wc[f * 2 * KERN_];
        float acc = 0.f;
#pragma unroll
        for (int k = 0; k < KERN_; ++k) acc += slab[0][tid + k] * w0[k];
#pragma unroll
        for (int k = 0; k < KERN_; ++k) acc += slab[1][tid + k] * w0[KERN_ + k];
        dst[f] = acc;
    }
}

// ---------------------------------------------------------------------------
// 2) q[b][a] = query[b,:] . Wq[a,:]
// ---------------------------------------------------------------------------
__global__ void query_proj_kernel(const float* __restrict__ query,
                                  const float* __restrict__ Wq,
                                  float* __restrict__ qb) {
    __shared__ float qrow[HID_];
    int b = blockIdx.x, a = threadIdx.x;
    for (int i = a; i < HID_; i += ATTN_) qrow[i] = query[b * HID_ + i];
    __syncthreads();
    const float* w = Wq + (size_t)a * HID_;
    float acc = 0.f;
    for (int h = 0; h < HID_; h += 4)
        acc += qrow[h] * w[h] + qrow[h + 1] * w[h + 1] +
               qrow[h + 2] * w[h + 2] + qrow[h + 3] * w[h + 3];
    qb[b * ATTN_ + a] = acc;
}

// ---------------------------------------------------------------------------
// 3) Pre-swizzle Wext = [Wv | Wloc] (ATTN x KTOT) into WMMA B-fragment order.
//    B (4x16, 32-bit): VGPR0 lanes0-15 = row K (N=lane), lanes16-31 = row K+2;
//    VGPR1 = rows K+1 / K+3.  Bfrag[kstep][atile][lane] = {v0, v1}.
// ---------------------------------------------------------------------------
__global__ void bfrag_kernel(const float* __restrict__ Wv,
                             const float* __restrict__ Wloc,
                             float* __restrict__ Bfrag) {
    int t = blockIdx.x * 256 + threadIdx.x;      // t = ks*256 + atile*32 + lane
    if (t >= KSTEPS * 8 * 32) return;
    int lane  = t & 31;
    int atile = (t >> 5) & 7;
    int ks    = t >> 8;
    int a = atile * 16 + (lane & 15);
    int k = ks * 4 + ((lane >> 4) << 1);
    float x0 = (k     < EMB_) ? Wv[a * EMB_ + k]     : Wloc[a * FILT_ + (k - EMB_)];
    float x1 = (k + 1 < EMB_) ? Wv[a * EMB_ + k + 1] : Wloc[a * FILT_ + (k + 1 - EMB_)];
    Bfrag[t * 2]     = x0;
    Bfrag[t * 2 + 1] = x1;
}

// ---------------------------------------------------------------------------
// 4) Fused GEMM + tanh + Wa reduction -> logits[b][v]
//    Per block: 64 v-rows x 128 attn, K = 544, 8 waves, 140 KB LDS.
// ---------------------------------------------------------------------------
__global__ void attn_logits_kernel(const float* __restrict__ value,
                                   const float* __restrict__ convT,
                                   const float* __restrict__ Bfrag,
                                   const float* __restrict__ qb,
                                   const float* __restrict__ bias,
                                   const float* __restrict__ Wa,
                                   const float* __restrict__ ba,
                                   float* __restrict__ logits) {
    __shared__ float X[MTILE * LDSS];            // 64 x 548 fp32 tile (value|conv)
    __shared__ float part[2][MTILE];             // per-attn-half logit partials
    int b  = blockIdx.y;
    int v0 = blockIdx.x * MTILE;
    int tid = threadIdx.x;

    // Stage extended-K tile into LDS (16B granularity, coalesced)
    const float* vsrc = value + ((size_t)b * V_ + v0) * EMB_;
    const float* csrc = convT + ((size_t)b * V_ + v0) * FILT_;
    for (int i = tid; i < MTILE * (KTOT / 4); i += 256) {
        int row = i / (KTOT / 4);
        int c4  = i % (KTOT / 4);
        const float* g = (c4 < EMB_ / 4)
                             ? vsrc + (size_t)row * EMB_ + c4 * 4
                             : csrc + (size_t)row * FILT_ + (c4 - EMB_ / 4) * 4;
        float* d = &X[row * LDSS + c4 * 4];
#ifdef USE_ASYNC_LDS
        __builtin_amdgcn_global_load_async_to_lds_b128(
            (gv4i_ptr)g, (lv4i_ptr)d, 0, 0);
#else
        float4 t = *(const float4*)g;
        d[0] = t.x; d[1] = t.y; d[2] = t.z; d[3] = t.w;
#endif
    }
#ifdef USE_ASYNC_LDS
    __builtin_amdgcn_s_wait_asynccnt(0);
#endif
    __syncthreads();

    int wave  = tid >> 5;
    int lane  = tid & 31;
    int m0    = (wave & 3) << 4;     // 16-row subtile
    int nhalf = wave >> 2;           // attn half (0: a 0-63, 1: a 64-127)
    int mrow  = lane & 15;
    int khi   = (lane >> 4) << 1;    // A frag: high half-wave holds K+2,K+3

    const float* xrow = &X[(m0 + mrow) * LDSS + khi];
    const v2f* bptr = (const v2f*)Bfrag + nhalf * 4 * 32 + lane;

    v8f acc[4] = {};
    for (int ks = 0; ks < KSTEPS; ++ks) {
        v2f afrag = *(const v2f*)(xrow + 4 * ks);     // ds_load_b64
        const v2f* bp = bptr + ks * 256;              // 8 atiles * 32 lanes
        acc[0] = __builtin_amdgcn_wmma_f32_16x16x4_f32(false, afrag, false, bp[0],  (short)0, acc[0], false, false);
        acc[1] = __builtin_amdgcn_wmma_f32_16x16x4_f32(false, afrag, false, bp[32], (short)0, acc[1], false, false);
        acc[2] = __builtin_amdgcn_wmma_f32_16x16x4_f32(false, afrag, false, bp[64], (short)0, acc[2], false, false);
        acc[3] = __builtin_amdgcn_wmma_f32_16x16x4_f32(false, afrag, false, bp[96], (short)0, acc[3], false, false);
    }

    // Epilogue: e = tanh(acc + q + bias); logit partial = sum_a e * Wa[a].
    // C/D layout: acc[j][r] is (M = m0 + r + 8*(lane>>4), N(attn) = a0j + (lane&15)).
    float rowpart[8];
#pragma unroll
    for (int r = 0; r < 8; ++r) rowpart[r] = 0.f;
    const float* qrow = qb + b * ATTN_;
#pragma unroll
    for (int j = 0; j < 4; ++j) {
        int a = ((nhalf * 4 + j) << 4) + (lane & 15);
        float qa  = qrow[a] + bias[a];
        float waj = Wa[a];
#pragma unroll
        for (int r = 0; r < 8; ++r)
            rowpart[r] += tanhf(acc[j][r] + qa) * waj;
    }
    // 16-lane half reduce; each wave owns rows [m0, m0+16) of its attn half:
    // lane 0 -> rows m0+r, lane 16 -> rows m0+8+r.  Race-free, deterministic.
#pragma unroll
    for (int r = 0; r < 8; ++r) {
        float s = rowpart[r];
        s += __shfl_xor(s, 1);
        s += __shfl_xor(s, 2);
        s += __shfl_xor(s, 4);
        s += __shfl_xor(s, 8);
        if ((lane & 15) == 0)
            part[nhalf][m0 + r + ((lane >> 4) << 3)] = s;
    }
    __syncthreads();
    if (tid < MTILE)
        logits[(size_t)b * V_ + v0 + tid] = part[0][tid] + part[1][tid] + ba[0];
}

// ---------------------------------------------------------------------------
// 5) Softmax over V (one block per batch row) -> alignment
// ---------------------------------------------------------------------------
__global__ void softmax_kernel(const float* __restrict__ logits,
                               float* __restrict__ align) {
    __shared__ float red[512];
    int b = blockIdx.x, tid = threadIdx.x;   // 512 threads
    const float* lrow = logits + (size_t)b * V_;
    float l0 = lrow[tid], l1 = lrow[tid + 512], l2 = lrow[tid + 1024], l3 = lrow[tid + 1536];

    red[tid] = fmaxf(fmaxf(l0, l1), fmaxf(l2, l3));
    __syncthreads();
    for (int s = 256; s > 0; s >>= 1) {
        if (tid < s) red[tid] = fmaxf(red[tid], red[tid + s]);
        __syncthreads();
    }
    float m = red[0];
    __syncthreads();

    float e0 = __expf(l0 - m), e1 = __expf(l1 - m), e2 = __expf(l2 - m), e3 = __expf(l3 - m);
    red[tid] = e0 + e1 + e2 + e3;
    __syncthreads();
    for (int s = 256; s > 0; s >>= 1) {
        if (tid < s) red[tid] += red[tid + s];
        __syncthreads();
    }
    float inv = 1.f / red[0];

    align[(size_t)b * V_ + tid]        = e0 * inv;
    align[(size_t)b * V_ + tid + 512]  = e1 * inv;
    align[(size_t)b * V_ + tid + 1024] = e2 * inv;
    align[(size_t)b * V_ + tid + 1536] = e3 * inv;
}

// ---------------------------------------------------------------------------
// 6) context partials: (32 v-chunks x B) blocks stream value once, coalesced
// ---------------------------------------------------------------------------
__global__ void ctx_partial_kernel(const float* __restrict__ value,
                                   const float* __restrict__ align,
                                   float* __restrict__ partials) {
    int c = blockIdx.x;                 // v-chunk of 64 rows
    int b = blockIdx.y;
    int e4 = threadIdx.x;               // 128 threads: one float4 emb column each
    const float* vb = value + ((size_t)b * V_ + c * 64) * EMB_ + e4 * 4;
    const float* al = align + (size_t)b * V_ + c * 64;
    float4 acc = make_float4(0.f, 0.f, 0.f, 0.f);
#pragma unroll 4
    for (int i = 0; i < 64; ++i) {
        float a = al[i];
        float4 t = *(const float4*)(vb + (size_t)i * EMB_);
        acc.x += a * t.x; acc.y += a * t.y; acc.z += a * t.z; acc.w += a * t.w;
    }
    *(float4*)(partials + ((size_t)c * B_ + b) * EMB_ + e4 * 4) = acc;
}

__global__ void ctx_reduce_kernel(const float* __restrict__ partials,
                                  float* __restrict__ ctx) {
    int b = blockIdx.x, e = threadIdx.x;      // 512 threads
    float s = 0.f;
#pragma unroll
    for (int c = 0; c < V_ / 64; ++c)
        s += partials[((size_t)c * B_ + b) * EMB_ + e];
    ctx[(size_t)b * EMB_ + e] = s;
}

// ---------------------------------------------------------------------------
extern "C" void kernel_launch(void* const* d_in, const int* in_sizes, int n_in,
                              void* d_out, int out_size, void* d_ws, size_t ws_size,
                              hipStream_t stream) {
    const float* query          = (const float*)d_in[0];
    const float* value          = (const float*)d_in[1];
    const float* last_alignment = (const float*)d_in[2];
    const float* Wq             = (const float*)d_in[3];
    const float* Wv             = (const float*)d_in[4];
    const float* Wa             = (const float*)d_in[5];
    const float* ba             = (const float*)d_in[6];
    const float* bias           = (const float*)d_in[7];
    const float* Wconv          = (const float*)d_in[8];
    const float* Wloc           = (const float*)d_in[9];

    float* ws       = (float*)d_ws;
    float* convT    = ws;                                    // B*V*FILT   = 4,194,304 f
    float* qb       = convT + (size_t)B_ * V_ * FILT_;       // B*ATTN     = 8,192 f
    float* Bfrag    = qb + B_ * ATTN_;                       // 136*8*32*2 = 69,632 f
    float* logits   = Bfrag + (size_t)KSTEPS * 8 * 32 * 2;   // B*V        = 131,072 f
    float* partials = logits + (size_t)B_ * V_;              // 32*B*EMB   = 1,048,576 f

    float* ctx   = (float*)d_out;            // (B, EMB)
    float* align = ctx + B_ * EMB_;          // (B, V)

    conv_loc_kernel<<<dim3(B_ * (V_ / 256)), 256, 0, stream>>>(last_alignment, Wconv, convT);
    query_proj_kernel<<<dim3(B_), ATTN_, 0, stream>>>(query, Wq, qb);
    bfrag_kernel<<<dim3(KSTEPS), 256, 0, stream>>>(Wv, Wloc, Bfrag);
    attn_logits_kernel<<<dim3(V_ / MTILE, B_), 256, 0, stream>>>(
        value, convT, Bfrag, qb, bias, Wa, ba, logits);
    softmax_kernel<<<dim3(B_), 512, 0, stream>>>(logits, align);
    ctx_partial_kernel<<<dim3(V_ / 64, B_), 128, 0, stream>>>(value, align, partials);
    ctx_reduce_kernel<<<dim3(B_), 512, 0, stream>>>(partials, ctx);
}